// EvolveGCNO_RecurrentGCN_45801531244828
// MI455X (gfx1250) — compile-verified
//
#include <hip/hip_runtime.h>
#include <math.h>

#define F 128
#define LOGC 48   // C=40 padded to 3 WMMA n-tiles

typedef __attribute__((ext_vector_type(2))) float v2f;
typedef __attribute__((ext_vector_type(8))) float v8f;

__device__ __forceinline__ v8f wmma_k4(v2f a, v2f b, v8f c) {
  // D = A(16x4) * B(4x16) + C(16x16), fp32 exact
  return __builtin_amdgcn_wmma_f32_16x16x4_f32(false, a, false, b, (short)0, c, false, false);
}

__device__ __forceinline__ float sigf(float x) { return 1.0f / (1.0f + __expf(-x)); }

// ---------------- 1) LSTM weight evolution: W = sig(o)*tanh(sig(i)*tanh(g)) ----------------
__global__ __launch_bounds__(256) void evolve_w(const float* __restrict__ W0,
                                                const float* __restrict__ Wih,
                                                const float* __restrict__ bih,
                                                const float* __restrict__ bhh,
                                                float* __restrict__ Wv) {
  int idx = blockIdx.x * 256 + threadIdx.x;   // 0 .. F*F-1
  int r = idx >> 7, j = idx & (F - 1);
  const float* w0r = W0 + (size_t)r * F;
  const float* wi = Wih + (size_t)j * F;            // i gate
  const float* wg = Wih + (size_t)(j + 2 * F) * F;  // g gate
  const float* wo = Wih + (size_t)(j + 3 * F) * F;  // o gate
  float gi = 0.f, gg = 0.f, go = 0.f;
  for (int k = 0; k < F; ++k) {
    float x0 = w0r[k];
    gi += x0 * wi[k]; gg += x0 * wg[k]; go += x0 * wo[k];
  }
  gi += bih[j] + bhh[j];
  gg += bih[j + 2 * F] + bhh[j + 2 * F];
  go += bih[j + 3 * F] + bhh[j + 3 * F];
  float c = sigf(gi) * tanhf(gg);          // c0 = 0, f-gate term vanishes
  Wv[idx] = sigf(go) * tanhf(c);
}

// ---------------- 2) degree / dinv ----------------
__global__ void deg_init(float* __restrict__ deg, int n) {
  int i = blockIdx.x * blockDim.x + threadIdx.x;
  if (i < n) deg[i] = 1.0f;                // self-loop weight 1
}
__global__ void deg_acc(const int* __restrict__ ei, const float* __restrict__ ew,
                        float* __restrict__ deg, int E_) {
  int e = blockIdx.x * blockDim.x + threadIdx.x;
  if (e < E_) unsafeAtomicAdd(&deg[ei[E_ + e]], ew[e]);   // segment over col
}
__global__ void deg_inv(float* __restrict__ deg, int n) {
  int i = blockIdx.x * blockDim.x + threadIdx.x;
  if (i < n) { float d = deg[i]; deg[i] = d > 0.f ? rsqrtf(d) : 0.f; }
}

// ---------------- 3) xw = x @ W  (WMMA f32 16x16x4, W staged in LDS) ----------------
// LDS layout: element (k, col) -> Ws[(k>>1)*256 + ((2*col) ^ ((kp&1)*32)) + (k&1)]
//  * B fragment {(k0,col),(k0+1,col)} becomes one contiguous ds_load_b64
//  * XOR-32 swizzle puts lane-halves (adjacent K-pair parities) in disjoint bank sets
__global__ __launch_bounds__(256) void xw_gemm(const float* __restrict__ x,
                                               const float* __restrict__ Wv,
                                               float* __restrict__ xw, int nrows) {
  __shared__ float Ws[F * F];              // 64 KB
  for (int i = threadIdx.x; i < F * F; i += 256) {
    int k = i >> 7, col = i & 127;
    int kp = k >> 1;
    Ws[kp * 256 + ((2 * col) ^ ((kp & 1) * 32)) + (k & 1)] = Wv[i];
  }
  __syncthreads();

  const int wave = threadIdx.x >> 5;
  const int lane = threadIdx.x & 31;
  const int m16 = lane & 15;
  const int kh2 = (lane >> 4) << 1;        // 0 or 2 (K sub-offset per lane half)
  const int hi8 = (lane >> 4) << 3;        // 0 or 8 (M offset for C/D)

  int rowBase = blockIdx.x * 128 + wave * 16;
  int row = rowBase + m16;
  int rowc = row < nrows ? row : (nrows - 1);   // clamp loads, keep EXEC full
  const float* xr = x + (size_t)rowc * F;

  v8f acc[8] = {};
  for (int kb = 0; kb < F; kb += 4) {
    int k0 = kb + kh2;
    v2f a = {xr[k0], xr[k0 + 1]};
    int kp = k0 >> 1;
    int swbit = kp & 1;
    const float* bbase = Ws + kp * 256 + 2 * m16;   // +(nt^swbit)*32 per tile
#pragma unroll
    for (int nt = 0; nt < 8; ++nt) {
      float2 bb = *(const float2*)(bbase + ((nt ^ swbit) << 5));
      v2f b = {bb.x, bb.y};
      acc[nt] = wmma_k4(a, b, acc[nt]);
    }
  }
#pragma unroll
  for (int nt = 0; nt < 8; ++nt)
#pragma unroll
    for (int r = 0; r < 8; ++r) {
      int rr = rowBase + r + hi8;
      if (rr < nrows) xw[(size_t)rr * F + nt * 16 + m16] = acc[nt][r];
    }
}

// ---------------- 4) h init with self-loop term: h = dinv^2 * xw ----------------
__global__ void h_self(const float* __restrict__ dinv, const float* __restrict__ xw,
                       float* __restrict__ h, int n) {
  int i = blockIdx.x * blockDim.x + threadIdx.x;
  if (i < n * F) {
    float di = dinv[i >> 7];
    h[i] = di * di * xw[i];
  }
}

// ---------------- 5) edge aggregation: h[col] += dinv[row]*ew*dinv[col] * xw[row] ----------------
__global__ __launch_bounds__(256) void edge_agg(const int* __restrict__ ei,
                                                const float* __restrict__ ew,
                                                const float* __restrict__ dinv,
                                                const float* __restrict__ xw,
                                                float* __restrict__ h, int E_) {
  int e = (blockIdx.x * 256 + threadIdx.x) >> 5;   // one wave per edge
  int lane = threadIdx.x & 31;
  if (e >= E_) return;
  int r = ei[e];            // gather source
  int c = ei[E_ + e];       // scatter target
  float nrm = dinv[r] * ew[e] * dinv[c];
  const float* src = xw + (size_t)r * F;
  float* dst = h + (size_t)c * F;
#pragma unroll
  for (int it = 0; it < 4; ++it) {
    int cc = it * 32 + lane;
    unsafeAtomicAdd(&dst[cc], nrm * src[cc]);
  }
}

// ---------------- 6) logits = relu(h) @ lin_W^T + b  (WMMA, C padded 40->48) ----------------
__global__ __launch_bounds__(256) void logits_gemm(const float* __restrict__ h,
                                                   const float* __restrict__ linW,
                                                   const float* __restrict__ linb,
                                                   float* __restrict__ logits, int nrows) {
  const int wave = threadIdx.x >> 5;
  const int lane = threadIdx.x & 31;
  const int m16 = lane & 15;
  const int kh2 = (lane >> 4) << 1;
  const int hi8 = (lane >> 4) << 3;

  int rowBase = blockIdx.x * 128 + wave * 16;
  int row = rowBase + m16;
  int rowc = row < nrows ? row : (nrows - 1);
  const float* hr = h + (size_t)rowc * F;

  v8f acc[3] = {};
  for (int kb = 0; kb < F; kb += 4) {
    int k0 = kb + kh2;
    v2f a = {fmaxf(hr[k0], 0.f), fmaxf(hr[k0 + 1], 0.f)};   // fused ReLU
#pragma unroll
    for (int nt = 0; nt < 3; ++nt) {
      int c = nt * 16 + m16;
      float2 wv = (c < 40) ? *(const float2*)(linW + (size_t)c * F + k0)
                           : make_float2(0.f, 0.f);          // B = lin_W^T, zero-pad cols
      v2f b = {wv.x, wv.y};
      acc[nt] = wmma_k4(a, b, acc[nt]);
    }
  }
#pragma unroll
  for (int nt = 0; nt < 3; ++nt) {
    int c = nt * 16 + m16;
    float bias = (c < 40) ? linb[c] : 0.f;
#pragma unroll
    for (int r = 0; r < 8; ++r) {
      int rr = rowBase + r + hi8;
      if (rr < nrows) logits[(size_t)rr * LOGC + c] = acc[nt][r] + bias;
    }
  }
}

// ---------------- 7) softmax over 40 classes, one wave per node ----------------
__global__ __launch_bounds__(256) void softmax40(const float* __restrict__ logits,
                                                 float* __restrict__ out, int nrows) {
  int nidx = (blockIdx.x * 256 + threadIdx.x) >> 5;
  int lane = threadIdx.x & 31;
  if (nidx >= nrows) return;
  const float* L = logits + (size_t)nidx * LOGC;
  float v0 = L[lane];                                   // lanes 0..31 -> cols 0..31 (<40)
  float v1 = (lane + 32 < 40) ? L[lane + 32] : -INFINITY;
  float m = fmaxf(v0, v1);
  for (int o = 16; o; o >>= 1) m = fmaxf(m, __shfl_xor(m, o, 32));
  float e0 = __expf(v0 - m);
  float e1 = (lane + 32 < 40) ? __expf(v1 - m) : 0.f;
  float s = e0 + e1;
  for (int o = 16; o; o >>= 1) s += __shfl_xor(s, o, 32);
  float inv = 1.0f / s;
  out[(size_t)nidx * 40 + lane] = e0 * inv;
  if (lane < 8) out[(size_t)nidx * 40 + 32 + lane] = e1 * inv;
}

extern "C" void kernel_launch(void* const* d_in, const int* in_sizes, int n_in,
                              void* d_out, int out_size, void* d_ws, size_t ws_size,
                              hipStream_t stream) {
  const float* x    = (const float*)d_in[0];
  const int*   ei   = (const int*)d_in[1];
  const float* ew   = (const float*)d_in[2];
  const float* W0   = (const float*)d_in[3];
  const float* Wih  = (const float*)d_in[4];
  // d_in[5] = W_hh: unused (h0 = 0 makes its term vanish)
  const float* bih  = (const float*)d_in[6];
  const float* bhh  = (const float*)d_in[7];
  const float* linW = (const float*)d_in[8];
  const float* linb = (const float*)d_in[9];
  float* out = (float*)d_out;

  const int N  = in_sizes[0] / F;     // 100000
  const int E_ = in_sizes[2];         // 600000

  char* ws = (char*)d_ws;
  float* Wv  = (float*)(ws);                                   // 64 KB
  float* deg = (float*)(ws + (64 << 10));                      // N*4
  float* xw  = (float*)(ws + (512 << 10));                     // N*F*4
  float* h   = (float*)(ws + (512 << 10) + (size_t)N * F * 4); // N*F*4
  float* logits = xw;  // xw is dead after edge_agg; reuse for N*48*4

  evolve_w<<<(F * F) / 256, 256, 0, stream>>>(W0, Wih, bih, bhh, Wv);
  deg_init<<<(N + 255) / 256, 256, 0, stream>>>(deg, N);
  deg_acc<<<(E_ + 255) / 256, 256, 0, stream>>>(ei, ew, deg, E_);
  deg_inv<<<(N + 255) / 256, 256, 0, stream>>>(deg, N);
  xw_gemm<<<(N + 127) / 128, 256, 0, stream>>>(x, Wv, xw, N);
  h_self<<<(N * F + 255) / 256, 256, 0, stream>>>(deg, xw, h, N);
  edge_agg<<<(E_ + 7) / 8, 256, 0, stream>>>(ei, ew, deg, xw, h, E_);
  logits_gemm<<<(N + 127) / 128, 256, 0, stream>>>(h, linW, linb, logits, N);
  softmax40<<<(N + 7) / 8, 256, 0, stream>>>(logits, out, N);
}